// MultivariateGaussianMixture_7499012899227
// MI455X (gfx1250) — compile-verified
//
#include <hip/hip_runtime.h>
#include <math.h>

typedef float v2f __attribute__((ext_vector_type(2)));
typedef float v8f __attribute__((ext_vector_type(8)));
typedef int   v4i __attribute__((ext_vector_type(4)));

#define N_PTS   32768
#define KC      32
#define DD      64
#define LOG_2PI 1.8378770664093453f

// --- gfx1250 async LDS-DMA path (guarded; falls back to sync staging) ------
#if defined(__has_builtin)
#  if __has_builtin(__builtin_amdgcn_global_load_async_to_lds_b128)
#    define GMM_ASYNC_LDS 1
#  endif
#  if __has_builtin(__builtin_amdgcn_s_wait_asynccnt)
#    define GMM_WAIT_ASYNC_BUILTIN 1
#  endif
#endif

// builtin expects int4 pointers qualified with the global/LDS address spaces
typedef __attribute__((address_space(1))) v4i gv4i;  // global (AS1)
typedef __attribute__((address_space(3))) v4i lv4i;  // LDS    (AS3)

__device__ __forceinline__ void gmm_wait_async0()
{
#if defined(GMM_ASYNC_LDS)
#  if defined(GMM_WAIT_ASYNC_BUILTIN)
    __builtin_amdgcn_s_wait_asynccnt(0);
#  else
    asm volatile("s_wait_asynccnt 0x0" ::: "memory");
#  endif
#endif
}

// Stage one 16KB component block (4096 floats) Wfrag[k] -> LDS buffer.
// 256 threads x 4 x b128: per-instruction a wave moves 512B fully coalesced.
__device__ __forceinline__ void gmm_stage(const float* __restrict__ src,
                                          float* dst, int tid)
{
#if defined(GMM_ASYNC_LDS)
#pragma unroll
    for (int i = 0; i < 4; ++i) {
        const int e = (tid + i * 256) * 4;
        __builtin_amdgcn_global_load_async_to_lds_b128(
            (gv4i*)(src + e), (lv4i*)(dst + e), /*offset=*/0, /*cpol=*/0);
    }
#else
#pragma unroll
    for (int i = 0; i < 4; ++i) {
        const int e = (tid + i * 256) * 4;
        *(float4*)&dst[e] = *(const float4*)&src[e];
    }
#endif
}

// ---------------------------------------------------------------------------
// Phase A: per-component prep. One block (64 threads) per component k.
//  Sigma = tril(L)tril(L)^T + I -> Cholesky C -> A = C^-1 -> b_k = A mu_k,
//  const_k = -0.5*(D log2pi + logdet) + log_softmax(w)_k.
//  A written pre-swizzled into WMMA-B fragment layout:
//   Wfrag[k][(jt*16+s)*64 + lane*2 + comp] = A[jt*16+(lane&15)][s*4+2*(lane>>4)+comp]
// ---------------------------------------------------------------------------
__global__ __launch_bounds__(64)
void gmm_prep(const float* __restrict__ L, const float* __restrict__ mu,
              const float* __restrict__ weights,
              float* __restrict__ Wfrag, float* __restrict__ bvec,
              float* __restrict__ constk)
{
    __shared__ float SS[64 * 65];   // Sigma / Cholesky factor (padded stride 65)
    __shared__ float CB[64 * 65];   // tril(L), later C^-1

    const int k = blockIdx.x;
    const int t = threadIdx.x;

    for (int c = 0; c < 64; ++c)
        CB[t * 65 + c] = (c <= t) ? L[(k * 64 + t) * 64 + c] : 0.0f;
    __syncthreads();

    // Sigma lower triangle
    for (int c = 0; c <= t; ++c) {
        float s = 0.0f;
        for (int d = 0; d <= c; ++d)
            s += CB[t * 65 + d] * CB[c * 65 + d];
        SS[t * 65 + c] = s + ((c == t) ? 1.0f : 0.0f);
    }
    __syncthreads();

    // in-place Cholesky (lower)
    for (int j = 0; j < 64; ++j) {
        if (t == j) SS[j * 65 + j] = sqrtf(SS[j * 65 + j]);
        __syncthreads();
        if (t > j) SS[t * 65 + j] /= SS[j * 65 + j];
        __syncthreads();
        if (t > j) {
            float f = SS[t * 65 + j];
            for (int c = j + 1; c <= t; ++c)
                SS[t * 65 + c] -= f * SS[c * 65 + j];
        }
        __syncthreads();
    }

    // triangular inverse: thread t computes column t of A = C^-1
    for (int c = 0; c < 64; ++c) CB[t * 65 + c] = 0.0f;
    __syncthreads();
    {
        CB[t * 65 + t] = 1.0f / SS[t * 65 + t];
        for (int i = t + 1; i < 64; ++i) {
            float s = 0.0f;
            for (int c = t; c < i; ++c)
                s += SS[i * 65 + c] * CB[c * 65 + t];
            CB[i * 65 + t] = -s / SS[i * 65 + i];
        }
    }
    __syncthreads();

    // b_k[t] = (A @ mu_k)[t]
    {
        float s = 0.0f;
        for (int c = 0; c <= t; ++c)
            s += CB[t * 65 + c] * mu[k * 64 + c];
        bvec[k * 64 + t] = s;
    }

    // const_k
    if (t == 0) {
        float logdet = 0.0f;
        for (int j = 0; j < 64; ++j) logdet += 2.0f * logf(SS[j * 65 + j]);
        float m = -INFINITY;
        for (int i = 0; i < KC; ++i) m = fmaxf(m, weights[i]);
        float se = 0.0f;
        for (int i = 0; i < KC; ++i) se += expf(weights[i] - m);
        float lw = weights[k] - (m + logf(se));
        constk[k] = -0.5f * (64.0f * LOG_2PI + logdet) + lw;
    }

    // write A in WMMA B-fragment layout
    for (int idx = t; idx < 4096; idx += 64) {
        int jts  = idx >> 6;            // jt*16 + s
        int l2   = idx & 63;            // lane*2 + comp
        int lane = l2 >> 1;
        int comp = l2 & 1;
        int jt   = jts >> 4;
        int s    = jts & 15;
        int row  = jt * 16 + (lane & 15);
        int col  = s * 4 + 2 * (lane >> 4) + comp;
        Wfrag[k * 4096 + idx] = CB[row * 65 + col];
    }
}

// ---------------------------------------------------------------------------
// Phase B: fused GEMM + Mahalanobis + logsumexp over components.
// 256 threads = 8 waves, 16 rows/wave. Double-buffered async LDS staging of
// each component's 16KB fragment block overlaps with the 64 WMMAs/wave of the
// previous component.
// ---------------------------------------------------------------------------
__global__ __launch_bounds__(256)
void gmm_maha(const float* __restrict__ X, const float* __restrict__ Wfrag,
              const float* __restrict__ bvec, const float* __restrict__ constk,
              float* __restrict__ loglik)
{
    __shared__ float buf[2][4096];   // double-buffered B fragments (2 x 16KB)
    __shared__ float bvLds[KC * 64]; // all b_k vectors (8KB)
    __shared__ float ckLds[KC];

    const int tid  = threadIdx.x;
    const int wave = tid >> 5;
    const int lane = tid & 31;
    const int m0   = blockIdx.x * 128 + wave * 16;
    const int row  = m0 + (lane & 15);
    const int h    = lane >> 4;   // which K-half this lane holds

    // preload b vectors + constants into LDS (published by first barrier)
    for (int i = tid; i < KC * 64; i += 256) bvLds[i] = bvec[i];
    if (tid < KC) ckLds[tid] = constk[tid];

    // prologue: kick off DMA of component 0
    gmm_stage(Wfrag, buf[0], tid);

    // A fragments: a[s] = X[row][s*4 + 2h .. +1]   (16 x v2f = 32 VGPRs)
    v2f a[16];
#pragma unroll
    for (int s = 0; s < 16; ++s)
        a[s] = *(const v2f*)&X[row * 64 + s * 4 + h * 2];

    // online logsumexp: lanes 0-15 track rows m0+0..7, lanes 16-31 rows m0+8..15
    float rmax[8], rsum[8];
#pragma unroll
    for (int r = 0; r < 8; ++r) { rmax[r] = -INFINITY; rsum[r] = 0.0f; }

    for (int k = 0; k < KC; ++k) {
        const int cur = k & 1;
        gmm_wait_async0();        // our slice of buf[cur] has landed
        __syncthreads();          // everyone's slice visible; buf[1-cur] free
        if (k + 1 < KC)
            gmm_stage(Wfrag + (k + 1) * 4096, buf[1 - cur], tid);
        const float* __restrict__ W = buf[cur];

        float maha[8] = {0, 0, 0, 0, 0, 0, 0, 0};
#pragma unroll
        for (int jt = 0; jt < 4; ++jt) {
            v8f c = 0.0f;
#pragma unroll
            for (int s = 0; s < 16; ++s) {
                v2f b = *(const v2f*)&W[(jt * 16 + s) * 64 + lane * 2];
                c = __builtin_amdgcn_wmma_f32_16x16x4_f32(
                        /*neg_a=*/false, a[s], /*neg_b=*/false, b,
                        /*c_mod=*/(short)0, c, /*reuse_a=*/false, /*reuse_b=*/false);
            }
            // lane's output column j = jt*16 + (lane&15); fold in -b_k[j], square
            float bv = bvLds[k * 64 + jt * 16 + (lane & 15)];
#pragma unroll
            for (int r = 0; r < 8; ++r) {
                float z = c[r] - bv;
                maha[r] = fmaf(z, z, maha[r]);
            }
        }
        // butterfly-sum across the 16 lanes of each half -> full maha per row
#pragma unroll
        for (int m = 1; m <= 8; m <<= 1)
#pragma unroll
            for (int r = 0; r < 8; ++r)
                maha[r] += __shfl_xor(maha[r], m, 32);

        // online logsumexp over components
        float ck = ckLds[k];
#pragma unroll
        for (int r = 0; r < 8; ++r) {
            float lp = ck - 0.5f * maha[r];
            float nm = fmaxf(rmax[r], lp);
            rsum[r] = rsum[r] * __expf(rmax[r] - nm) + __expf(lp - nm);
            rmax[r] = nm;
        }
    }

    if (lane == 0 || lane == 16) {
        int base = m0 + (h ? 8 : 0);
#pragma unroll
        for (int r = 0; r < 8; ++r)
            loglik[base + r] = rmax[r] + __logf(rsum[r]);
    }
}

// ---------------------------------------------------------------------------
// Phase C: out = -logsumexp_n(loglik)   (single block, stable two-pass)
// ---------------------------------------------------------------------------
__global__ __launch_bounds__(1024)
void gmm_final(const float* __restrict__ loglik, float* __restrict__ out)
{
    __shared__ float red[1024];
    const int t = threadIdx.x;

    float m = -INFINITY;
    for (int i = t; i < N_PTS; i += 1024) m = fmaxf(m, loglik[i]);
    red[t] = m;
    __syncthreads();
    for (int s = 512; s > 0; s >>= 1) {
        if (t < s) red[t] = fmaxf(red[t], red[t + s]);
        __syncthreads();
    }
    float gmax = red[0];
    __syncthreads();

    float s = 0.0f;
    for (int i = t; i < N_PTS; i += 1024) s += __expf(loglik[i] - gmax);
    red[t] = s;
    __syncthreads();
    for (int st = 512; st > 0; st >>= 1) {
        if (t < st) red[t] += red[t + st];
        __syncthreads();
    }
    if (t == 0) out[0] = -(gmax + __logf(red[0]));
}

// ---------------------------------------------------------------------------
extern "C" void kernel_launch(void* const* d_in, const int* in_sizes, int n_in,
                              void* d_out, int out_size, void* d_ws, size_t ws_size,
                              hipStream_t stream)
{
    const float* X  = (const float*)d_in[0];   // [N, 64]
    const float* mu = (const float*)d_in[1];   // [K, 64]
    const float* L  = (const float*)d_in[2];   // [K, 64, 64]
    const float* w  = (const float*)d_in[3];   // [K]
    // d_in[4] = it (unused)

    float* Wfrag  = (float*)d_ws;              // KC*4096 floats (512 KB)
    float* bvec   = Wfrag + KC * 4096;         // KC*64
    float* constk = bvec + KC * 64;            // KC
    float* loglik = constk + KC;               // N_PTS floats

    gmm_prep <<<KC, 64, 0, stream>>>(L, mu, w, Wfrag, bvec, constk);
    gmm_maha <<<N_PTS / 128, 256, 0, stream>>>(X, Wfrag, bvec, constk, loglik);
    gmm_final<<<1, 1024, 0, stream>>>(loglik, (float*)d_out);
}